// TripletLoss_17257178595659
// MI455X (gfx1250) — compile-verified
//
#include <hip/hip_runtime.h>
#include <math.h>

// ---------------------------------------------------------------------------
// Triplet softplus loss on MI455X (gfx1250, wave32).
//
// Strategy: never build the 8192x8192 distance matrix. For each group of 16
// triplets, gather rows into f32 WMMA fragments and use V_WMMA_F32_16X16X4_F32
// (full f32 precision -> matches the f32 reference) to compute
// diag(A_i * B_j^T) and diag(A_i * B_k^T), i.e. the 2x16 needed dot products.
// Features (16 MB) are L2-resident on a 192 MB L2 chip, so gathered traffic
// stays on-die; HBM sees only ~19 MB per call.
// ---------------------------------------------------------------------------

typedef __attribute__((ext_vector_type(2))) float v2f;
typedef __attribute__((ext_vector_type(8))) float v8f;

#define D_DIM 512

// --- Kernel 1: squared row norms, one wave (32 lanes) per row ---------------
__global__ void sqnorm_kernel(const float* __restrict__ F,
                              float* __restrict__ sq, int N) {
    const int lane = threadIdx.x & 31;
    const int wave = threadIdx.x >> 5;
    const int row  = blockIdx.x * (blockDim.x >> 5) + wave;
    if (row >= N) return;                      // wave-uniform guard, no barriers
    const float4* p = (const float4*)(F + (size_t)row * D_DIM);
    float acc = 0.0f;
#pragma unroll
    for (int s = 0; s < 4; ++s) {              // 512 floats = 128 float4, 32 lanes x 4
        float4 v = p[lane + 32 * s];
        acc += v.x * v.x + v.y * v.y + v.z * v.z + v.w * v.w;
    }
#pragma unroll
    for (int off = 16; off > 0; off >>= 1)
        acc += __shfl_down(acc, off, 32);
    if (lane == 0) sq[row] = acc;
}

// Select element idx (0..7) of a v8f accumulator with a cndmask tree.
__device__ __forceinline__ float sel8(v8f c, int idx) {
    float a0 = (idx & 1) ? c[1] : c[0];
    float a1 = (idx & 1) ? c[3] : c[2];
    float a2 = (idx & 1) ? c[5] : c[4];
    float a3 = (idx & 1) ? c[7] : c[6];
    float b0 = (idx & 2) ? a1 : a0;
    float b1 = (idx & 2) ? a3 : a2;
    return (idx & 4) ? b1 : b0;
}

// --- Kernel 2: 16 triplets per wave via f32 WMMA ----------------------------
__global__ void triplet_wmma_kernel(const float* __restrict__ F,
                                    const int* __restrict__ trip,
                                    const float* __restrict__ sq,
                                    float* __restrict__ partials,
                                    int T, int numTiles) {
    const int lane          = threadIdx.x & 31;
    const int wave          = threadIdx.x >> 5;
    const int wavesPerBlock = blockDim.x >> 5;

    const int tileOrig = blockIdx.x * wavesPerBlock + wave;
    const int tile     = (tileOrig < numTiles) ? tileOrig : (numTiles - 1);

    const int r    = lane & 15;   // triplet-in-tile == M row == N col (diagonal)
    const int half = lane >> 4;   // K sub-block selector for f32 16x16x4 frags

    const int  t     = tile * 16 + r;
    const bool valid = (t < T) && (tileOrig < numTiles) && (lane < 16);
    const int  tc    = (t < T) ? t : (T - 1);

    const int ir = trip[3 * tc + 0];
    const int jr = trip[3 * tc + 1];
    const int kr = trip[3 * tc + 2];

    // ISA layout for V_WMMA_F32_16X16X4_F32:
    //   A (16x4):  lane = {M = lane%16, K = 2*half + v},   v in {0,1}
    //   B (4x16):  lane = {N = lane%16, K = 2*half + v}    (same gather pattern,
    //              since B[k][n] = x_j[n][k])
    const float* Ap  = F + (size_t)ir * D_DIM + half * 2;
    const float* Bjp = F + (size_t)jr * D_DIM + half * 2;
    const float* Bkp = F + (size_t)kr * D_DIM + half * 2;

    v8f cj = {};
    v8f ck = {};
#pragma unroll 4
    for (int kk = 0; kk < D_DIM; kk += 4) {
        v2f a  = *(const v2f*)(Ap  + kk);
        v2f bj = *(const v2f*)(Bjp + kk);
        v2f bk = *(const v2f*)(Bkp + kk);
        cj = __builtin_amdgcn_wmma_f32_16x16x4_f32(false, a, false, bj,
                                                   (short)0, cj, false, false);
        ck = __builtin_amdgcn_wmma_f32_16x16x4_f32(false, a, false, bk,
                                                   (short)0, ck, false, false);
    }

    // Diagonal extraction. C layout: VGPR v, lane l holds D[v + 8*(l/16)][l%16].
    // D[t][t] lives in lane t / vgpr t (t<8) and lane t+16 / vgpr t-8 (t>=8).
    const int   idx  = lane & 7;
    const float selJ = sel8(cj, idx);
    const float selK = sel8(ck, idx);
    const int   src  = (r < 8) ? r : (r + 16);
    const float dij  = __shfl(selJ, src, 32);
    const float dik  = __shfl(selK, src, 32);

    float partial = 0.0f;
    if (valid) {
        const float si = sq[ir], sj = sq[jr], sk = sq[kr];
        const float dj = fmaxf(si + sj - 2.0f * dij, 0.0f);  // clip(d_ij, 0)
        const float dk = fmaxf(si + sk - 2.0f * dik, 0.0f);  // clip(d_ik, 0)
        const float z  = dj - dk;
        // numerically stable softplus
        partial = fmaxf(z, 0.0f) + log1pf(expf(-fabsf(z)));
    }

#pragma unroll
    for (int off = 16; off > 0; off >>= 1)
        partial += __shfl_down(partial, off, 32);

    __shared__ float wsum[8];
    if (lane == 0) wsum[wave] = partial;
    __syncthreads();
    if (threadIdx.x == 0) {
        float s = 0.0f;
        for (int w = 0; w < wavesPerBlock; ++w) s += wsum[w];
        partials[blockIdx.x] = s;
    }
}

// --- Kernel 3: deterministic final reduction --------------------------------
__global__ void reduce_kernel(const float* __restrict__ partials, int n,
                              float* __restrict__ out, float invT) {
    float acc = 0.0f;
    for (int i = threadIdx.x; i < n; i += blockDim.x) acc += partials[i];
    const int lane = threadIdx.x & 31;
    const int wave = threadIdx.x >> 5;
#pragma unroll
    for (int off = 16; off > 0; off >>= 1)
        acc += __shfl_down(acc, off, 32);
    __shared__ float wsum[8];
    if (lane == 0) wsum[wave] = acc;
    __syncthreads();
    if (threadIdx.x == 0) {
        float s = 0.0f;
        for (int w = 0; w < (int)(blockDim.x >> 5); ++w) s += wsum[w];
        out[0] = s * invT;
    }
}

extern "C" void kernel_launch(void* const* d_in, const int* in_sizes, int n_in,
                              void* d_out, int out_size, void* d_ws, size_t ws_size,
                              hipStream_t stream) {
    (void)n_in; (void)out_size; (void)ws_size;
    const float* F    = (const float*)d_in[0];
    const int*   trip = (const int*)d_in[1];

    const int N = in_sizes[0] / D_DIM;   // 8192
    const int T = in_sizes[1] / 3;       // 262144

    float* sq       = (float*)d_ws;      // N floats
    float* partials = sq + N;            // one per block of kernel 2

    // Kernel 1: 8 waves/block, one row per wave.
    const int sqBlocks = (N + 7) / 8;
    sqnorm_kernel<<<sqBlocks, 256, 0, stream>>>(F, sq, N);

    // Kernel 2: 16 triplets per wave, 8 waves per block.
    const int numTiles = (T + 15) / 16;
    const int blocks   = (numTiles + 7) / 8;
    triplet_wmma_kernel<<<blocks, 256, 0, stream>>>(F, trip, sq, partials,
                                                    T, numTiles);

    // Kernel 3: single block, deterministic sum.
    reduce_kernel<<<1, 256, 0, stream>>>(partials, blocks, (float*)d_out,
                                         1.0f / (float)T);
}